// LambdaRankLoss_27049704031075
// MI455X (gfx1250) — compile-verified
//
#include <hip/hip_runtime.h>
#include <math.h>

#define KNEG  32
#define DFULL 129
#define VSZ   2048
#define NCAND 33

typedef __attribute__((ext_vector_type(2))) float v2f;
typedef __attribute__((ext_vector_type(8))) float v8f;

__device__ __forceinline__ float wave_sum(float v) {
#pragma unroll
  for (int m = 16; m > 0; m >>= 1) v += __shfl_xor(v, m, 32);
  return v;
}
__device__ __forceinline__ float wave_max(float v) {
#pragma unroll
  for (int m = 16; m > 0; m >>= 1) v = fmaxf(v, __shfl_xor(v, m, 32));
  return v;
}

// One wave32 per query. WMMA f32 16x16x4 computes the 32 anchor·negative
// spatial dot products (dims 1..128); time term and positive handled in VALU.
__global__ __launch_bounds__(32)
void lambdarank_per_query(const float* __restrict__ anchor_emb,
                          const float* __restrict__ positive_emb,
                          const float* __restrict__ negative_embs,
                          const float* __restrict__ tree,
                          const int* __restrict__ anchor_idx,
                          const int* __restrict__ pos_idx,
                          const int* __restrict__ neg_idx,
                          float* __restrict__ loss_ws) {
  const int q    = blockIdx.x;
  const int lane = threadIdx.x;
  const int h    = lane >> 4;   // K half within a 4-chunk
  const int r    = lane & 15;   // column (candidate-in-tile)

  const float* a  = anchor_emb   + (size_t)q * DFULL;
  const float* p  = positive_emb + (size_t)q * DFULL;
  const float* ng = negative_embs + (size_t)q * KNEG * DFULL;

  // ---- WMMA: spatial dots anchor · neg_{0..31} over dims 1..128 ----
  // A: 16x4, all rows = anchor chunk. Documented layout: lane<16 holds K0,K1
  // in v0,v1; lane>=16 holds K2,K3. B: 4x16, lane = column N, same K mapping.
  const float* row0 = ng + (size_t)r * DFULL;         // tile 0: neg 0..15
  const float* row1 = ng + (size_t)(r + 16) * DFULL;  // tile 1: neg 16..31
  v8f acc0 = {};
  v8f acc1 = {};
#pragma unroll 4
  for (int c = 0; c < 32; ++c) {
    const int k0 = 1 + 4 * c + 2 * h;
    v2f A;  A.x  = a[k0];     A.y  = a[k0 + 1];
    v2f B0; B0.x = row0[k0];  B0.y = row0[k0 + 1];
    v2f B1; B1.x = row1[k0];  B1.y = row1[k0 + 1];
    acc0 = __builtin_amdgcn_wmma_f32_16x16x4_f32(false, A, false, B0,
                                                 (short)0, acc0, false, false);
    acc1 = __builtin_amdgcn_wmma_f32_16x16x4_f32(false, A, false, B1,
                                                 (short)0, acc1, false, false);
  }
  // All rows of D identical (A rows identical); lane l holds dot for neg_l.
  float sdot = (lane < 16) ? acc0[0] : acc1[0];

  // ---- positive distance (exact fp32, butterfly reduction) ----
  float part = 0.f;
  for (int i = lane; i < DFULL; i += 32) {
    float s = (i == 0) ? -1.f : 1.f;
    part += s * a[i] * p[i];
  }
  float inner_pos = wave_sum(part);
  float dpos = acoshf(fmaxf(-inner_pos, 1.0f + 1e-7f));

  // ---- negative distances: inner = -a0*c0 + sdot ----
  float a0 = a[0];
  float c0 = ng[(size_t)lane * DFULL];
  float dneg = acoshf(fmaxf(a0 * c0 - sdot, 1.0f + 1e-7f));

  // ---- tree-distance relevances ----
  int ai = anchor_idx[q];
  const float* trow = tree + (size_t)ai * VSZ;
  float tpos = trow[pos_idx[q]];
  float tneg = trow[neg_idx[(size_t)q * KNEG + lane]];
  float maxt = fmaxf(wave_max(tneg), tpos);
  float rpos = (maxt - tpos + 1e-6f) / (maxt + 1e-6f);
  float rneg = (maxt - tneg + 1e-6f) / (maxt + 1e-6f);

  // ---- stage per-candidate arrays (cand 0 = positive, 1..32 = negatives) ----
  __shared__ float dA[NCAND], rA[NCAND], gA[NCAND];
  dA[lane + 1] = dneg;
  rA[lane + 1] = rneg;
  if (lane == 0) { dA[0] = dpos; rA[0] = rpos; }
  __syncthreads();

  // ---- ranks (stable argsort-of-argsort), disc, and IDCG ----
  const int ncand = (lane == 31) ? 2 : 1;  // lane 31 also covers candidate 32
  float partIdcg = 0.f;
  for (int s = 0; s < ncand; ++s) {
    int i = (s == 0) ? lane : 32;
    float di = dA[i], ri = rA[i];
    int cntd = 0, cntr = 0;
    for (int j = 0; j < NCAND; ++j) {
      float dj = dA[j], rj = rA[j];
      cntd += (int)((dj < di) || (dj == di && j < i));  // ascending rank of d
      cntr += (int)((rj > ri) || (rj == ri && j < i));  // descending rank of rel
    }
    int rank = cntd + 1;
    float disc = (rank <= 10) ? (1.0f / log2f((float)rank + 1.0f)) : 0.0f;
    gA[i] = disc;
    float w = (cntr < 10) ? (1.0f / log2f((float)cntr + 2.0f)) : 0.0f;
    partIdcg += ri * w;
  }
  __syncthreads();
  float idcg = wave_sum(partIdcg);
  float inv_idcg = (idcg > 0.f) ? (1.0f / idcg) : 0.0f;

  // ---- pairwise lambda sum ----
  float partLoss = 0.f;
  for (int s = 0; s < ncand; ++s) {
    int i = (s == 0) ? lane : 32;
    float di = dA[i], ri = rA[i], gi = gA[i];
    for (int j = 0; j < NCAND; ++j) {
      float drel  = ri - rA[j];
      float ddisc = gi - gA[j];
      float S = (drel > 0.f) ? 1.f : ((drel < 0.f) ? -1.f : 0.f);
      float delta = fabsf(drel * ddisc) * inv_idcg;
      float dd = dA[j] - di;                       // dd[i][j] = d_j - d_i
      float sig = 1.0f / (1.0f + expf(-(S * dd))); // SIGMA = 1
      partLoss += (S * delta * sig) * (-dd);
    }
  }
  float tot = wave_sum(partLoss);
  if (lane == 0) loss_ws[q] = 0.5f * tot;
}

// Deterministic fixed-order reduction of per-query losses.
__global__ __launch_bounds__(256)
void final_reduce(const float* __restrict__ ws, float* __restrict__ out, int Bn) {
  __shared__ float sm[256];
  int t = threadIdx.x;
  float acc = 0.f;
  for (int i = t; i < Bn; i += 256) acc += ws[i];
  sm[t] = acc;
  __syncthreads();
  for (int off = 128; off > 0; off >>= 1) {
    if (t < off) sm[t] += sm[t + off];
    __syncthreads();
  }
  if (t == 0) out[0] = 0.15f * (sm[0] / (float)Bn);
}

extern "C" void kernel_launch(void* const* d_in, const int* in_sizes, int n_in,
                              void* d_out, int out_size, void* d_ws, size_t ws_size,
                              hipStream_t stream) {
  (void)n_in; (void)out_size; (void)ws_size;
  const float* anchor = (const float*)d_in[0];
  const float* pos    = (const float*)d_in[1];
  const float* negs   = (const float*)d_in[2];
  const float* tree   = (const float*)d_in[3];
  const int*   aidx   = (const int*)d_in[4];
  const int*   pidx   = (const int*)d_in[5];
  const int*   nidx   = (const int*)d_in[6];
  float* ws  = (float*)d_ws;
  float* out = (float*)d_out;
  const int Bn = in_sizes[4];  // number of queries (anchor_idx count)

  lambdarank_per_query<<<Bn, 32, 0, stream>>>(anchor, pos, negs, tree,
                                              aidx, pidx, nidx, ws);
  final_reduce<<<1, 256, 0, stream>>>(ws, out, Bn);
}